// GCN_Mutag_2250562863403
// MI455X (gfx1250) — compile-verified
//
#include <hip/hip_runtime.h>
#include <math.h>

typedef __attribute__((ext_vector_type(2))) float v2f;
typedef __attribute__((ext_vector_type(8))) float v8f;

#define GCN_N 100000
#define GCN_E 800000
#define GCN_F 128
#define GCN_H 64
#define GCN_G 512

// -----------------------------------------------------------------------------
// C[N x 64] = A[N x K] @ B[K x 64]   (row-major, K in {128, 64}, N % 16 == 0)
// One wave -> one 16x16 output tile via V_WMMA_F32_16X16X4_F32.
// Block = 128 threads = 4 waves = one 16-row stripe covering all 64 columns.
// VGPR layouts per CDNA5 ISA 7.12.2:
//   A 16x4 : lane L (m=L%16, hi=L/16): a = {A[m][k0+2*hi], A[m][k0+2*hi+1]}
//   B 4x16 : lane L (n=L%16, hi=L/16): b = {B[k0+2*hi][n], B[k0+2*hi+1][n]}
//   D 16x16: vgpr i, lanes 0-15 -> M=i, lanes 16-31 -> M=i+8, N=lane%16
// -----------------------------------------------------------------------------
__global__ __launch_bounds__(128) void gemm_wmma_f32(
    const float* __restrict__ A, const float* __restrict__ B,
    float* __restrict__ C, int K)
{
    const int lane = threadIdx.x & 31;
    const int wave = threadIdx.x >> 5;          // 0..3 -> 16-col tile
    const int row0 = blockIdx.x * 16;
    const int col0 = wave * 16;

    const int m  = lane & 15;
    const int hi = lane >> 4;                    // 0 or 1

    v8f acc = {};
    const float* ap = A + (size_t)(row0 + m) * K + 2 * hi;
    const float* bp = B + (size_t)(2 * hi) * GCN_H + col0 + m;

    for (int k0 = 0; k0 < K; k0 += 4) {
        v2f a, b;
        a.x = ap[k0 + 0];
        a.y = ap[k0 + 1];
        b.x = bp[(size_t)k0 * GCN_H];
        b.y = bp[(size_t)(k0 + 1) * GCN_H];
        acc = __builtin_amdgcn_wmma_f32_16x16x4_f32(
            /*neg_a=*/false, a, /*neg_b=*/false, b,
            /*c_mod=*/(short)0, acc, /*reuse_a=*/false, /*reuse_b=*/false);
    }

    float* cp = C + (size_t)row0 * GCN_H + col0 + m;
#pragma unroll
    for (int i = 0; i < 8; ++i) {
        const int M = i + 8 * hi;
        cp[(size_t)M * GCN_H] = acc[i];
    }
}

// -----------------------------------------------------------------------------
// agg[rows[e]] += support[cols[e]]  -- one wave per edge, float2 per lane.
// -----------------------------------------------------------------------------
__global__ __launch_bounds__(256) void edge_scatter(
    const int* __restrict__ rows, const int* __restrict__ cols,
    const float* __restrict__ support, float* __restrict__ agg, int E)
{
    const int t = blockIdx.x * blockDim.x + threadIdx.x;
    const int e = t >> 5;
    if (e >= E) return;
    const int lane = t & 31;
    const int r = rows[e];
    const int c = cols[e];
    const float2 v = *(const float2*)(support + (size_t)c * GCN_H + lane * 2);
    float* dst = agg + (size_t)r * GCN_H + lane * 2;
    atomicAdd(dst + 0, v.x);
    atomicAdd(dst + 1, v.y);
}

// -----------------------------------------------------------------------------
// out = (relu?) (agg + bias)
// -----------------------------------------------------------------------------
__global__ __launch_bounds__(256) void bias_act(
    const float* __restrict__ agg, const float* __restrict__ bias,
    float* __restrict__ out, int total, int do_relu)
{
    const int t = blockIdx.x * blockDim.x + threadIdx.x;
    if (t >= total) return;
    const int h = t & (GCN_H - 1);
    float v = agg[t] + bias[h];
    if (do_relu) v = fmaxf(v, 0.0f);
    out[t] = v;
}

// -----------------------------------------------------------------------------
// Per-graph sum + count: one wave per node.
// -----------------------------------------------------------------------------
__global__ __launch_bounds__(256) void pool_scatter(
    const float* __restrict__ x, const int* __restrict__ batch,
    float* __restrict__ sums, float* __restrict__ counts, int n)
{
    const int t = blockIdx.x * blockDim.x + threadIdx.x;
    const int node = t >> 5;
    if (node >= n) return;
    const int lane = t & 31;
    const int g = batch[node];
    const float2 v = *(const float2*)(x + (size_t)node * GCN_H + lane * 2);
    float* dst = sums + (size_t)g * GCN_H + lane * 2;
    atomicAdd(dst + 0, v.x);
    atomicAdd(dst + 1, v.y);
    if (lane == 0) atomicAdd(&counts[g], 1.0f);
}

// -----------------------------------------------------------------------------
// Head: mean -> 64x16 relu -> 16x8 relu -> 8x1 sigmoid. One thread per graph.
// -----------------------------------------------------------------------------
__global__ __launch_bounds__(64) void head_mlp(
    const float* __restrict__ sums, const float* __restrict__ counts,
    const float* __restrict__ dw1, const float* __restrict__ db1,
    const float* __restrict__ dw2, const float* __restrict__ db2,
    const float* __restrict__ dw3, const float* __restrict__ db3,
    float* __restrict__ out, int G)
{
    const int g = blockIdx.x * blockDim.x + threadIdx.x;
    if (g >= G) return;
    const float inv_cnt = 1.0f / fmaxf(counts[g], 1.0f);

    float mean[GCN_H];
#pragma unroll
    for (int i = 0; i < GCN_H; ++i) mean[i] = sums[(size_t)g * GCN_H + i] * inv_cnt;

    float h1[16];
    for (int j = 0; j < 16; ++j) {
        float s = db1[j];
        for (int i = 0; i < GCN_H; ++i) s += mean[i] * dw1[i * 16 + j];
        h1[j] = fmaxf(s, 0.0f);
    }
    float h2[8];
    for (int j = 0; j < 8; ++j) {
        float s = db2[j];
        for (int i = 0; i < 16; ++i) s += h1[i] * dw2[i * 8 + j];
        h2[j] = fmaxf(s, 0.0f);
    }
    float s = db3[0];
    for (int i = 0; i < 8; ++i) s += h2[i] * dw3[i];
    out[g] = 1.0f / (1.0f + expf(-s));
}

// -----------------------------------------------------------------------------
extern "C" void kernel_launch(void* const* d_in, const int* in_sizes, int n_in,
                              void* d_out, int out_size, void* d_ws, size_t ws_size,
                              hipStream_t stream)
{
    const float* feat  = (const float*)d_in[0];   // [N, F]
    const int*   edge  = (const int*)  d_in[1];   // [2, E]
    const int*   batch = (const int*)  d_in[2];   // [N]
    const float* w1 = (const float*)d_in[3];  const float* b1 = (const float*)d_in[4];
    const float* w2 = (const float*)d_in[5];  const float* b2 = (const float*)d_in[6];
    const float* w3 = (const float*)d_in[7];  const float* b3 = (const float*)d_in[8];
    const float* dw1 = (const float*)d_in[9];  const float* db1 = (const float*)d_in[10];
    const float* dw2 = (const float*)d_in[11]; const float* db2 = (const float*)d_in[12];
    const float* dw3 = (const float*)d_in[13]; const float* db3 = (const float*)d_in[14];

    const int N = GCN_N, E = GCN_E, F = GCN_F, H = GCN_H, G = GCN_G;
    const int* rows = edge;        // edge_index[0]
    const int* cols = edge + E;    // edge_index[1]

    // Workspace layout (floats)
    float* support = (float*)d_ws;                     // N*H
    float* agg     = support + (size_t)N * H;          // N*H
    float* x       = agg     + (size_t)N * H;          // N*H
    float* sums    = x       + (size_t)N * H;          // G*H
    float* counts  = sums    + (size_t)G * H;          // G

    const int gemm_grid    = N / 16;                   // N % 16 == 0
    const int scatter_grid = (E * 32 + 255) / 256;
    const int ew_grid      = (N * H + 255) / 256;
    const int pool_grid    = (N * 32 + 255) / 256;

    // ---- Layer 1: K = F = 128, relu
    gemm_wmma_f32<<<gemm_grid, 128, 0, stream>>>(feat, w1, support, F);
    hipMemsetAsync(agg, 0, (size_t)N * H * sizeof(float), stream);
    edge_scatter<<<scatter_grid, 256, 0, stream>>>(rows, cols, support, agg, E);
    bias_act<<<ew_grid, 256, 0, stream>>>(agg, b1, x, N * H, 1);

    // ---- Layer 2: K = H = 64, relu
    gemm_wmma_f32<<<gemm_grid, 128, 0, stream>>>(x, w2, support, H);
    hipMemsetAsync(agg, 0, (size_t)N * H * sizeof(float), stream);
    edge_scatter<<<scatter_grid, 256, 0, stream>>>(rows, cols, support, agg, E);
    bias_act<<<ew_grid, 256, 0, stream>>>(agg, b2, x, N * H, 1);

    // ---- Layer 3: K = H = 64, no relu
    gemm_wmma_f32<<<gemm_grid, 128, 0, stream>>>(x, w3, support, H);
    hipMemsetAsync(agg, 0, (size_t)N * H * sizeof(float), stream);
    edge_scatter<<<scatter_grid, 256, 0, stream>>>(rows, cols, support, agg, E);
    bias_act<<<ew_grid, 256, 0, stream>>>(agg, b3, x, N * H, 0);

    // ---- Global mean pool + head
    hipMemsetAsync(sums, 0, (size_t)(G * H + G) * sizeof(float), stream);
    pool_scatter<<<pool_grid, 256, 0, stream>>>(x, batch, sums, counts, N);
    head_mlp<<<(G + 63) / 64, 64, 0, stream>>>(sums, counts, dw1, db1, dw2, db2,
                                               dw3, db3, (float*)d_out, G);
}